// Attention_30202210025502
// MI455X (gfx1250) — compile-verified
//
#include <hip/hip_runtime.h>
#include <stdint.h>

// Attention context kernel for MI455X (gfx1250), compile-only loop.
//
// reference: energy = enc·w_enc + const(b); p = softmax_s(energy);
//            context[b,e] = sum_s p[s] * enc[b,s,e]
// The dec_hidden/bias terms are constant over s -> softmax-invariant -> dropped.
//
// Design: memory-bound (512 MiB enc, ~0.5 FLOP/B). Single HBM pass with online
// (flash) softmax. HBM->LDS via CDNA5 async loads (ASYNCcnt, double buffer).
// Context rank-16 updates via V_WMMA_F32_16X16X4_F32 out of LDS.

#define ENC_E 1024   // 2*ENC_HID
#define SEQ   2048
#define BATCH 64
#define TROWS 16     // s-rows per LDS tile
#define ROWB  4096   // bytes per row (1024 f32)

typedef __attribute__((ext_vector_type(2))) float v2f;
typedef __attribute__((ext_vector_type(4))) float v4f;
typedef __attribute__((ext_vector_type(8))) float v8f;

// Generic pointers to LDS carry the LDS byte offset in their low 32 bits
// (flat LDS aperture; addr[31:0] is the wave-relative LDS address).
__device__ __forceinline__ unsigned lds_off(const void* p) {
  return (unsigned)(uintptr_t)p;
}

__device__ __forceinline__ void async_tile_load(unsigned lds_base, unsigned gvoff_base,
                                                unsigned long long saddr, unsigned tid) {
  // 16 rows x 4096B tile; thread t moves 16B of every row -> 16 async b128/thread.
  const unsigned lt = lds_base + tid * 16u;
  const unsigned gt = gvoff_base + tid * 16u;
#pragma unroll
  for (int i = 0; i < 16; ++i) {
    unsigned la = lt + (unsigned)i * 4096u;
    unsigned va = gt + (unsigned)i * 4096u;
    asm volatile("global_load_async_to_lds_b128 %0, %1, %2 offset:0"
                 :: "v"(la), "v"(va), "s"(saddr) : "memory");
  }
}

__global__ __launch_bounds__(256)
void attn_partial(const float* __restrict__ enc, const float* __restrict__ attw,
                  float* __restrict__ outp, float* __restrict__ ml,
                  int NCH, int KS, int finalize)
{
  __shared__ float sbuf[2 * TROWS * ENC_E];   // 2 x 64KB tile buffers
  __shared__ float e_arr[TROWS];
  __shared__ float p_arr[TROWS];
  __shared__ float sm_m, sm_l, sm_scale;

  const int tid  = threadIdx.x;
  const int lane = tid & 31;
  const int wave = tid >> 5;
  const int b = blockIdx.x / NCH;
  const int c = blockIdx.x % NCH;

  if (tid == 0) { sm_m = -3.0e38f; sm_l = 0.0f; sm_scale = 0.0f; }

  // per-lane w_enc registers: lane covers float4 chunks k*32+lane, k=0..7
  v4f wreg[8];
#pragma unroll
  for (int k = 0; k < 8; ++k) wreg[k] = ((const v4f*)attw)[k * 32 + lane];

  // WMMA context accumulators: each wave owns 8 n-tiles of 16 e-columns
  v8f acc[8] = {};

  const int T = KS / TROWS;
  const unsigned long long gbase =
      (unsigned long long)(uintptr_t)(enc + ((size_t)b * SEQ + (size_t)c * KS) * ENC_E);
  const unsigned lbuf[2] = { lds_off(&sbuf[0]), lds_off(&sbuf[TROWS * ENC_E]) };

  // prologue: tile 0 -> buf0
  async_tile_load(lbuf[0], 0u, gbase, tid);

  for (int t = 0; t < T; ++t) {
    // issue next tile into the other buffer, then retire tile t's loads
    if (t + 1 < T) {
      async_tile_load(lbuf[(t + 1) & 1], (unsigned)(t + 1) * 65536u, gbase, tid);
      asm volatile("s_wait_asynccnt 0x10" ::: "memory");   // <=16 outstanding (next tile)
    } else {
      asm volatile("s_wait_asynccnt 0x0" ::: "memory");
    }
    __syncthreads();   // tile t visible to all waves

    const float* buf = &sbuf[(t & 1) * TROWS * ENC_E];

    // ---- energy: wave w computes rows 2w, 2w+1 (VALU FMA + wave32 reduce) ----
#pragma unroll
    for (int j = 0; j < 2; ++j) {
      const int r = wave * 2 + j;
      const v4f* row = (const v4f*)(buf + r * ENC_E);
      float part = 0.0f;
#pragma unroll
      for (int k = 0; k < 8; ++k) {
        v4f v = row[k * 32 + lane];
        part += v.x * wreg[k].x + v.y * wreg[k].y + v.z * wreg[k].z + v.w * wreg[k].w;
      }
#pragma unroll
      for (int off = 16; off > 0; off >>= 1) part += __shfl_xor(part, off);
      if (lane == 0) e_arr[r] = part;
    }
    __syncthreads();

    // ---- online softmax bookkeeping (wave 0) ----
    if (wave == 0) {
      float ev  = (lane < 16) ? e_arr[lane] : -3.0e38f;
      float red = ev;
#pragma unroll
      for (int off = 8; off > 0; off >>= 1) red = fmaxf(red, __shfl_xor(red, off));
      float tmax  = __shfl(red, 0);
      float m_old = sm_m, l_old = sm_l;
      float m_new = fmaxf(m_old, tmax);
      float scl   = __expf(m_old - m_new);
      float p     = (lane < 16) ? __expf(ev - m_new) : 0.0f;
      if (lane < 16) p_arr[lane] = p;
      float ps = p;
#pragma unroll
      for (int off = 16; off > 0; off >>= 1) ps += __shfl_xor(ps, off);
      if (lane == 0) { sm_m = m_new; sm_l = l_old * scl + ps; sm_scale = scl; }
    }
    __syncthreads();

    // ---- context rank-16 update via V_WMMA_F32_16X16X4_F32 ----
    // A[m,k] = p[s_k] (broadcast over M); B[k,n] = enc[s_k, e0+n]; D=C accumulate.
    {
      const float scl = sm_scale;
      const int hi2  = (lane & 16) ? 2 : 0;   // K pair held by this half-wave
      const int ncol = lane & 15;
#pragma unroll
      for (int nt = 0; nt < 8; ++nt) acc[nt] *= scl;   // flash rescale
#pragma unroll
      for (int nt = 0; nt < 8; ++nt) {
        const int e0 = (wave * 8 + nt) * 16;
#pragma unroll
        for (int ks = 0; ks < 4; ++ks) {               // 4 K-groups of 4 s-rows
          const int srow = ks * 4 + hi2;
          v2f a = *(const v2f*)&p_arr[srow];           // {p[K], p[K+1]}
          v2f bm;
          bm.x = buf[srow * ENC_E + e0 + ncol];
          bm.y = buf[(srow + 1) * ENC_E + e0 + ncol];
          acc[nt] = __builtin_amdgcn_wmma_f32_16x16x4_f32(
              false, a, false, bm, (short)0, acc[nt], false, false);
        }
      }
    }
    __syncthreads();   // all reads of buf[t&1] done before it is reloaded
  }

  // ---- emit partial (or final) result; D row M=0 lives in VGPR0 lanes 0..15 ----
  const float l   = sm_l;
  const float m   = sm_m;
  const float inv = finalize ? (1.0f / l) : 1.0f;
  const size_t obase = (size_t)blockIdx.x * ENC_E;
  if (lane < 16) {
#pragma unroll
    for (int nt = 0; nt < 8; ++nt)
      outp[obase + (size_t)(wave * 8 + nt) * 16 + lane] = acc[nt][0] * inv;
  }
  if (!finalize && tid == 0) { ml[2 * blockIdx.x] = m; ml[2 * blockIdx.x + 1] = l; }
}

// flash-merge of per-chunk partials: ctx = sum_i e^(m_i-M) acc_i / sum_i e^(m_i-M) l_i
__global__ __launch_bounds__(256)
void attn_merge(const float* __restrict__ pctx, const float* __restrict__ ml,
                float* __restrict__ out, int NCH)
{
  const int b = blockIdx.x, t = threadIdx.x;
  float M = -3.0e38f;
  for (int i = 0; i < NCH; ++i) M = fmaxf(M, ml[2 * (b * NCH + i)]);
  float L = 0.0f;
  for (int i = 0; i < NCH; ++i)
    L += ml[2 * (b * NCH + i) + 1] * __expf(ml[2 * (b * NCH + i)] - M);
  const float invL = 1.0f / L;
  v4f s = {0.0f, 0.0f, 0.0f, 0.0f};
  for (int i = 0; i < NCH; ++i) {
    const float f = __expf(ml[2 * (b * NCH + i)] - M);
    v4f a = ((const v4f*)(pctx + (size_t)(b * NCH + i) * ENC_E))[t];
    s += a * f;
  }
  ((v4f*)(out + (size_t)b * ENC_E))[t] = s * invL;
}

extern "C" void kernel_launch(void* const* d_in, const int* in_sizes, int n_in,
                              void* d_out, int out_size, void* d_ws, size_t ws_size,
                              hipStream_t stream) {
  (void)in_sizes; (void)n_in; (void)out_size;
  const float* enc  = (const float*)d_in[0];   // (64,2048,1024) f32
  const float* attw = (const float*)d_in[2];   // (1,2048) f32; first 1024 = w_enc
  float* out = (float*)d_out;                  // (64,1024) f32

  int NCH = 8;                                 // S-chunks per batch (occupancy)
  auto need = [](int n) -> size_t { return (size_t)BATCH * n * (ENC_E + 2) * sizeof(float); };
  while (NCH > 1 && need(NCH) > ws_size) NCH >>= 1;
  const int KS = SEQ / NCH;

  if (NCH == 1) {
    attn_partial<<<BATCH, 256, 0, stream>>>(enc, attw, out, nullptr, 1, SEQ, 1);
  } else {
    float* pctx = (float*)d_ws;
    float* ml   = pctx + (size_t)BATCH * NCH * ENC_E;
    attn_partial<<<BATCH * NCH, 256, 0, stream>>>(enc, attw, pctx, ml, NCH, KS, 0);
    attn_merge<<<BATCH, 256, 0, stream>>>(pctx, ml, out, NCH);
  }
}